// LiteMLA_79559974191572
// MI455X (gfx1250) — compile-verified
//
#include <hip/hip_runtime.h>
#include <hip/hip_bf16.h>

typedef __attribute__((ext_vector_type(2))) float v2f;
typedef __attribute__((ext_vector_type(8))) float v8f;
typedef int v4i_ __attribute__((vector_size(16)));   // b128 payload type

#define NPIX 4096        // 64*64
#define CIN  256
#define QKVC 768
#define PROJ 512

#if defined(__HIP_DEVICE_COMPILE__) && \
    __has_builtin(__builtin_amdgcn_global_load_async_to_lds_b128) && \
    __has_builtin(__builtin_amdgcn_s_wait_asynccnt)
#define USE_ASYNC 1
#else
#define USE_ASYNC 0
#endif

#define GLOBAL_AS __attribute__((address_space(1)))
#define LDS_AS    __attribute__((address_space(3)))

// ---------------------------------------------------------------------------
// 16B global -> LDS copy. Async DMA path on CDNA5 (ASYNCcnt), VGPR path else.
// ---------------------------------------------------------------------------
__device__ __forceinline__ void copy16_g2l(const float* g, float* l) {
#if USE_ASYNC
    __builtin_amdgcn_global_load_async_to_lds_b128(
        (GLOBAL_AS v4i_*)(void*)g, (LDS_AS v4i_*)(void*)l, 0, 0);
#else
    *reinterpret_cast<float4*>(l) = *reinterpret_cast<const float4*>(g);
#endif
}

__device__ __forceinline__ void copy16_wait() {
#if USE_ASYNC
    __builtin_amdgcn_s_wait_asynccnt(0);
#endif
}

// ---------------------------------------------------------------------------
// WMMA helper: D = A(16x4) * B(4x16) + C, full fp32
// ---------------------------------------------------------------------------
__device__ __forceinline__ v8f wmma4(v2f a, v2f b, v8f c) {
    return __builtin_amdgcn_wmma_f32_16x16x4_f32(false, a, false, b, (short)0, c,
                                                 false, false);
}

// ---------------------------------------------------------------------------
// Kernel 1: qkv = W(768x256) * X(256x4096) per batch.  WMMA f32 + async LDS
// double buffering.  block = 256 thr (8 waves); block tile 128co x 64n.
// grid: (64, 6, 16)
// ---------------------------------------------------------------------------
__global__ __launch_bounds__(256) void k_qkv(const float* __restrict__ x,
                                             const float* __restrict__ w,
                                             float* __restrict__ y) {
    __shared__ float lds[2][16 * 64];
    const int b    = blockIdx.z;
    const int n0   = blockIdx.x * 64;
    const int tid  = threadIdx.x;
    const int lane = tid & 31;
    const int m    = lane & 15;
    const int hi   = (lane < 16) ? 0 : 8;
    const int kh   = (lane < 16) ? 0 : 2;
    const int co0  = blockIdx.y * 128 + (tid >> 5) * 16;

    const float* xb = x + (size_t)b * CIN * NPIX;
    float*       yb = y + (size_t)b * QKVC * NPIX;

    v8f acc[4] = {};

    const int ldr = tid >> 4;            // 0..15  (row within K-chunk)
    const int ldc = (tid & 15) << 2;     // 0..60  (col, float4)

    // preload chunk 0
    copy16_g2l(xb + (size_t)ldr * NPIX + n0 + ldc, &lds[0][ldr * 64 + ldc]);

    for (int kc = 0; kc < CIN; kc += 16) {
        const int cur = (kc >> 4) & 1;
        copy16_wait();
        __syncthreads();                 // chunk visible; prev readers done
        if (kc + 16 < CIN)               // start next chunk into other buffer
            copy16_g2l(xb + (size_t)(kc + 16 + ldr) * NPIX + n0 + ldc,
                       &lds[cur ^ 1][ldr * 64 + ldc]);

#pragma unroll
        for (int kk = 0; kk < 16; kk += 4) {
            v2f a;
            a.x = w[(size_t)(co0 + m) * CIN + kc + kk + kh];
            a.y = w[(size_t)(co0 + m) * CIN + kc + kk + kh + 1];
#pragma unroll
            for (int t = 0; t < 4; ++t) {
                v2f bf;
                bf.x = lds[cur][(kk + kh) * 64 + t * 16 + m];
                bf.y = lds[cur][(kk + kh + 1) * 64 + t * 16 + m];
                acc[t] = wmma4(a, bf, acc[t]);
            }
        }
        __syncthreads();                 // readers done before next overwrite
    }

#pragma unroll
    for (int t = 0; t < 4; ++t)
#pragma unroll
        for (int r = 0; r < 8; ++r)
            yb[(size_t)(co0 + r + hi) * NPIX + n0 + t * 16 + m] = acc[t][r];
}

// ---------------------------------------------------------------------------
// Kernel 2: depthwise 5x5, pad 2.  One thread per output element.
// ---------------------------------------------------------------------------
__global__ __launch_bounds__(256) void k_dw(const float* __restrict__ in,
                                            const float* __restrict__ wdw,
                                            float* __restrict__ out) {
    const long long idx = (long long)blockIdx.x * 256 + threadIdx.x;
    const long long total = (long long)16 * QKVC * NPIX;
    if (idx >= total) return;
    const int p  = (int)(idx & (NPIX - 1));
    const int c  = (int)((idx >> 12) % QKVC);
    const int hh = p >> 6, ww = p & 63;
    const float* src = in + (idx - p);   // start of this (b,c) plane
    const float* wt  = wdw + c * 25;
    float s = 0.f;
#pragma unroll
    for (int dy = 0; dy < 5; ++dy) {
        const int yy = hh + dy - 2;
        if (yy < 0 || yy >= 64) continue;
#pragma unroll
        for (int dx = 0; dx < 5; ++dx) {
            const int xx = ww + dx - 2;
            if (xx < 0 || xx >= 64) continue;
            s += src[yy * 64 + xx] * wt[dy * 5 + dx];
        }
    }
    out[idx] = s;
}

// ---------------------------------------------------------------------------
// Kernel 3: grouped pointwise (48 groups, 16in->16out), WMMA, in-place.
// block 256 thr; wave w -> 4 tiles of 16 cols.  grid: (8, 48, 16)
// ---------------------------------------------------------------------------
__global__ __launch_bounds__(256) void k_pw(float* __restrict__ buf,
                                            const float* __restrict__ wpw) {
    const int b    = blockIdx.z;
    const int g    = blockIdx.y;
    const int lane = threadIdx.x & 31;
    const int wv   = threadIdx.x >> 5;
    const int m    = lane & 15;
    const int hi   = (lane < 16) ? 0 : 8;
    const int kh   = (lane < 16) ? 0 : 2;

    float* base = buf + (size_t)b * QKVC * NPIX + (size_t)g * 16 * NPIX;
    const float* wg = wpw + g * 16 * 16;   // [o][i]

    v2f a[4];
#pragma unroll
    for (int s = 0; s < 4; ++s) {
        a[s].x = wg[m * 16 + s * 4 + kh];
        a[s].y = wg[m * 16 + s * 4 + kh + 1];
    }

    const int n0 = blockIdx.x * 512 + wv * 64;
#pragma unroll
    for (int t = 0; t < 4; ++t) {
        const int col = n0 + t * 16;
        v8f acc = {};
#pragma unroll
        for (int s = 0; s < 4; ++s) {
            v2f bf;
            bf.x = base[(size_t)(s * 4 + kh) * NPIX + col + m];
            bf.y = base[(size_t)(s * 4 + kh + 1) * NPIX + col + m];
            acc = wmma4(a[s], bf, acc);
        }
#pragma unroll
        for (int r = 0; r < 8; ++r)
            base[(size_t)(r + hi) * NPIX + col + m] = acc[r];   // reads done
    }
}

// ---------------------------------------------------------------------------
// Kernel 4: vk[b,h] = v_pad(17xN) * k_relu^T(Nx16).
// 8 waves split N=4096; WMMA chain over K; LDS reduce.  grid: (32, 16)
// ---------------------------------------------------------------------------
__global__ __launch_bounds__(256) void k_vk(const float* __restrict__ qkv,
                                            const float* __restrict__ agg,
                                            float* __restrict__ vk) {
    __shared__ float red[8 * 32 * 8];
    __shared__ float kred[8 * 32];
    const int ht = blockIdx.x, b = blockIdx.y;
    const float* msp = (ht < 16)
        ? qkv + ((size_t)b * QKVC + (size_t)ht * 48) * NPIX
        : agg + ((size_t)b * QKVC + (size_t)(ht - 16) * 48) * NPIX;

    const int lane = threadIdx.x & 31;
    const int wv   = threadIdx.x >> 5;
    const int m    = lane & 15;
    const int kh   = (lane < 16) ? 0 : 2;

    const float* vrow = msp + (size_t)(32 + m) * NPIX;   // A row d=m
    const float* krow = msp + (size_t)(16 + m) * NPIX;   // B col e=m

    v8f acc = {};
    float ks = 0.f;
    const int nbeg = wv * 512, nend = nbeg + 512;
    for (int n = nbeg; n < nend; n += 4) {
        v2f a, bf;
        a.x  = vrow[n + kh];
        a.y  = vrow[n + kh + 1];
        bf.x = fmaxf(krow[n + kh], 0.f);
        bf.y = fmaxf(krow[n + kh + 1], 0.f);
        ks  += bf.x + bf.y;                 // relu(k) row-sum contribution
        acc  = wmma4(a, bf, acc);
    }

#pragma unroll
    for (int r = 0; r < 8; ++r) red[(wv * 32 + lane) * 8 + r] = acc[r];
    kred[wv * 32 + lane] = ks;
    __syncthreads();

    if (wv == 0) {
        float s[8];
#pragma unroll
        for (int r = 0; r < 8; ++r) s[r] = 0.f;
        for (int w2 = 0; w2 < 8; ++w2)
#pragma unroll
            for (int r = 0; r < 8; ++r) s[r] += red[(w2 * 32 + lane) * 8 + r];
        float* vkb = vk + ((size_t)b * 32 + ht) * 272;   // 17*16
        const int hi = (lane < 16) ? 0 : 8;
#pragma unroll
        for (int r = 0; r < 8; ++r) vkb[(r + hi) * 16 + m] = s[r];
        if (lane < 16) {
            float t = 0.f;
            for (int w2 = 0; w2 < 8; ++w2)
                t += kred[w2 * 32 + lane] + kred[w2 * 32 + lane + 16];
            vkb[256 + lane] = t;
        }
    }
}

// ---------------------------------------------------------------------------
// Kernel 5: out = (vk(16x16) * relu(q)(16xN)) / clip(den, eps).
// grid: (32, 32, 16); block 256; wave -> one 16-col tile, block covers 128.
// ---------------------------------------------------------------------------
__global__ __launch_bounds__(256) void k_apply(const float* __restrict__ qkv,
                                               const float* __restrict__ agg,
                                               const float* __restrict__ vk,
                                               float* __restrict__ attn) {
    __shared__ float vks[272];
    const int ht = blockIdx.y, b = blockIdx.z;
    const float* msp = (ht < 16)
        ? qkv + ((size_t)b * QKVC + (size_t)ht * 48) * NPIX
        : agg + ((size_t)b * QKVC + (size_t)(ht - 16) * 48) * NPIX;
    const float* vkb = vk + ((size_t)b * 32 + ht) * 272;
    for (int i = threadIdx.x; i < 272; i += 256) vks[i] = vkb[i];
    __syncthreads();

    const int lane = threadIdx.x & 31;
    const int wv   = threadIdx.x >> 5;
    const int m    = lane & 15;
    const int hi   = (lane < 16) ? 0 : 8;
    const int kh   = (lane < 16) ? 0 : 2;
    const int n0   = blockIdx.x * 128 + wv * 16;

    v8f acc = {};
#pragma unroll
    for (int s = 0; s < 4; ++s) {
        v2f a, bf;
        a.x  = vks[m * 16 + s * 4 + kh];
        a.y  = vks[m * 16 + s * 4 + kh + 1];
        bf.x = fmaxf(msp[(size_t)(s * 4 + kh) * NPIX + n0 + m], 0.f);
        bf.y = fmaxf(msp[(size_t)(s * 4 + kh + 1) * NPIX + n0 + m], 0.f);
        acc  = wmma4(a, bf, acc);
    }

    float den = 0.f;
#pragma unroll
    for (int e = 0; e < 16; ++e)
        den += vks[256 + e] * fmaxf(msp[(size_t)e * NPIX + n0 + m], 0.f);
    const float inv = 1.0f / fmaxf(den, 1e-15f);

    float* ab = attn + ((size_t)b * PROJ + (size_t)ht * 16) * NPIX;
#pragma unroll
    for (int r = 0; r < 8; ++r)
        ab[(size_t)(r + hi) * NPIX + n0 + m] = acc[r] * inv;
}

// ---------------------------------------------------------------------------
// Kernel 6: out = BN( Wp(256x512) * attn(512x4096) ) per batch.
// grid: (64, 2, 16); block 256 (8 waves); async LDS double buffering.
// ---------------------------------------------------------------------------
__global__ __launch_bounds__(256) void k_proj(const float* __restrict__ attn,
                                              const float* __restrict__ w,
                                              const float* __restrict__ gamma,
                                              const float* __restrict__ beta,
                                              const float* __restrict__ mean,
                                              const float* __restrict__ var,
                                              float* __restrict__ out) {
    __shared__ float lds[2][16 * 64];
    const int b    = blockIdx.z;
    const int n0   = blockIdx.x * 64;
    const int tid  = threadIdx.x;
    const int lane = tid & 31;
    const int m    = lane & 15;
    const int hi   = (lane < 16) ? 0 : 8;
    const int kh   = (lane < 16) ? 0 : 2;
    const int co0  = blockIdx.y * 128 + (tid >> 5) * 16;

    const float* xb = attn + (size_t)b * PROJ * NPIX;
    float*       yb = out + (size_t)b * CIN * NPIX;

    v8f acc[4] = {};
    const int ldr = tid >> 4;
    const int ldc = (tid & 15) << 2;

    copy16_g2l(xb + (size_t)ldr * NPIX + n0 + ldc, &lds[0][ldr * 64 + ldc]);

    for (int kc = 0; kc < PROJ; kc += 16) {
        const int cur = (kc >> 4) & 1;
        copy16_wait();
        __syncthreads();
        if (kc + 16 < PROJ)
            copy16_g2l(xb + (size_t)(kc + 16 + ldr) * NPIX + n0 + ldc,
                       &lds[cur ^ 1][ldr * 64 + ldc]);

#pragma unroll
        for (int kk = 0; kk < 16; kk += 4) {
            v2f a;
            a.x = w[(size_t)(co0 + m) * PROJ + kc + kk + kh];
            a.y = w[(size_t)(co0 + m) * PROJ + kc + kk + kh + 1];
#pragma unroll
            for (int t = 0; t < 4; ++t) {
                v2f bf;
                bf.x = lds[cur][(kk + kh) * 64 + t * 16 + m];
                bf.y = lds[cur][(kk + kh + 1) * 64 + t * 16 + m];
                acc[t] = wmma4(a, bf, acc[t]);
            }
        }
        __syncthreads();
    }

#pragma unroll
    for (int r = 0; r < 8; ++r) {
        const int co = co0 + r + hi;
        const float iv = gamma[co] * rsqrtf(var[co] + 1e-5f);
        const float bb = beta[co] - mean[co] * iv;
#pragma unroll
        for (int t = 0; t < 4; ++t)
            yb[(size_t)co * NPIX + n0 + t * 16 + m] = acc[t][r] * iv + bb;
    }
}

// ---------------------------------------------------------------------------
extern "C" void kernel_launch(void* const* d_in, const int* in_sizes, int n_in,
                              void* d_out, int out_size, void* d_ws, size_t ws_size,
                              hipStream_t stream) {
    const float* x      = (const float*)d_in[0];
    const float* w_qkv  = (const float*)d_in[1];
    const float* w_dw   = (const float*)d_in[2];
    const float* w_pw   = (const float*)d_in[3];
    const float* w_proj = (const float*)d_in[4];
    const float* bn_g   = (const float*)d_in[5];
    const float* bn_b   = (const float*)d_in[6];
    const float* bn_m   = (const float*)d_in[7];
    const float* bn_v   = (const float*)d_in[8];
    float* out = (float*)d_out;

    // workspace carve-up (floats)
    float* qkv  = (float*)d_ws;                            // 16*768*4096
    float* dw   = qkv  + (size_t)16 * QKVC * NPIX;         // 16*768*4096 (also agg, in-place)
    float* attn = dw   + (size_t)16 * QKVC * NPIX;         // 16*512*4096
    float* vkb  = attn + (size_t)16 * PROJ * NPIX;         // 16*32*272

    k_qkv<<<dim3(64, 6, 16), 256, 0, stream>>>(x, w_qkv, qkv);

    const long long dwtot = (long long)16 * QKVC * NPIX;
    k_dw<<<(unsigned)((dwtot + 255) / 256), 256, 0, stream>>>(qkv, w_dw, dw);

    k_pw<<<dim3(8, 48, 16), 256, 0, stream>>>(dw, w_pw);   // in-place -> agg

    k_vk<<<dim3(32, 16), 256, 0, stream>>>(qkv, dw, vkb);

    k_apply<<<dim3(32, 32, 16), 256, 0, stream>>>(qkv, dw, vkb, attn);

    k_proj<<<dim3(64, 2, 16), 256, 0, stream>>>(attn, w_proj, bn_g, bn_b, bn_m,
                                                bn_v, out);
    (void)in_sizes; (void)n_in; (void)out_size; (void)ws_size;
}